// CamMemory_23519240913060
// MI455X (gfx1250) — compile-verified
//
#include <hip/hip_runtime.h>
#include <hip/hip_bf16.h>

#define B_SZ 64
#define D_SZ 2048
#define N_SZ 16384
#define INV_TEMP 20.0f

#define BK 32
#define LDSS 36   // row stride (floats): multiple of 4 -> 16B-aligned rows for async b128
                  // writes; 36*r mod 64 hits 16 distinct banks for r=0..15 -> conflict-free

typedef float v2f __attribute__((ext_vector_type(2)));
typedef float v8f __attribute__((ext_vector_type(8)));

// Flat (generic) pointer to a __shared__ object: low 32 bits are the wave-relative
// LDS byte offset (ISA 10.2: LDS aperture keeps the offset in addr[31:0]).
__device__ __forceinline__ unsigned lds_off(const void* p) {
    return (unsigned)(size_t)p;
}

// ---------------- Kernel 1: x = inputs / ||inputs||  (per row) ----------------
__global__ void __launch_bounds__(256)
normalize_rows(const float* __restrict__ in, float* __restrict__ xn) {
    __shared__ float red[256];
    const int row = blockIdx.x;
    const int t = threadIdx.x;
    const float* r = in + (size_t)row * D_SZ;
    float s = 0.f;
    #pragma unroll
    for (int i = 0; i < D_SZ / 256; ++i) {
        float v = r[t + i * 256];
        s += v * v;
    }
    red[t] = s;
    __syncthreads();
    for (int off = 128; off > 0; off >>= 1) {
        if (t < off) red[t] += red[t + off];
        __syncthreads();
    }
    const float rn = rsqrtf(red[0]);
    float* o = xn + (size_t)row * D_SZ;
    #pragma unroll
    for (int i = 0; i < D_SZ / 256; ++i)
        o[t + i * 256] = r[t + i * 256] * rn;
}

// ---------------- Kernel 2: logits = (xn @ features^T) * (1/TEMP) ----------------
// Block: 256 threads = 8 waves. Block tile: 16 (M) x 128 (N), K chunked by 32.
// Double-buffered LDS filled by async global->LDS DMA (ASYNCcnt), overlapped with
// V_WMMA_F32_16X16X4_F32 compute on the previous chunk.
__global__ void __launch_bounds__(256)
gemm_wmma(const float* __restrict__ xn, const float* __restrict__ feat,
          float* __restrict__ logits) {
    __shared__ float lsA[2][16 * LDSS];    //  16 x 32 tile of xn     (double-buffered)
    __shared__ float lsB[2][128 * LDSS];   // 128 x 32 tile of feats  (double-buffered)

    const int t    = threadIdx.x;
    const int lane = t & 31;
    const int wave = t >> 5;                 // 0..7
    const int n0   = blockIdx.x * 128;       // column tile origin (N)
    const int m0   = blockIdx.y * 16;        // row tile origin (M)

    // staging maps: every thread issues exactly 5 async loads per chunk
    const int rowA = t >> 4;                 // 0..15  (b64: 2 floats)
    const int colA = (t & 15) * 2;           // 0..30
    const int rowB = t >> 3;                 // 0..31 (+i*32)  (b128: 4 floats)
    const int colB = (t & 7) * 4;            // 0..28

    // ISA 7.12.2 f32 fragment addressing
    const int lm   = lane & 15;              // A: M row / B: N row within tile
    const int koff = (lane >> 4) * 2;        // lower half K+0..1, upper half K+2..3

    const float* gA  = xn   + (size_t)(m0 + rowA)      * D_SZ + colA;
    const float* gB0 = feat + (size_t)(n0 + rowB)      * D_SZ + colB;
    const float* gB1 = feat + (size_t)(n0 + rowB + 32) * D_SZ + colB;
    const float* gB2 = feat + (size_t)(n0 + rowB + 64) * D_SZ + colB;
    const float* gB3 = feat + (size_t)(n0 + rowB + 96) * D_SZ + colB;

    const unsigned laA  = lds_off(&lsA[0][rowA * LDSS + colA]);
    const unsigned laB0 = lds_off(&lsB[0][(rowB)      * LDSS + colB]);
    const unsigned laB1 = lds_off(&lsB[0][(rowB + 32) * LDSS + colB]);
    const unsigned laB2 = lds_off(&lsB[0][(rowB + 64) * LDSS + colB]);
    const unsigned laB3 = lds_off(&lsB[0][(rowB + 96) * LDSS + colB]);
    const unsigned bufStrideA = sizeof(float) * 16 * LDSS;
    const unsigned bufStrideB = sizeof(float) * 128 * LDSS;

    // issue one K-chunk's async global->LDS copies (5 per thread, in order)
    auto issue = [&](int k0, int buf) {
        const unsigned oa = buf ? bufStrideA : 0u;
        const unsigned ob = buf ? bufStrideB : 0u;
        asm volatile("global_load_async_to_lds_b64 %0, %1, off"
                     :: "v"(laA + oa), "v"(gA + k0) : "memory");
        asm volatile("global_load_async_to_lds_b128 %0, %1, off"
                     :: "v"(laB0 + ob), "v"(gB0 + k0) : "memory");
        asm volatile("global_load_async_to_lds_b128 %0, %1, off"
                     :: "v"(laB1 + ob), "v"(gB1 + k0) : "memory");
        asm volatile("global_load_async_to_lds_b128 %0, %1, off"
                     :: "v"(laB2 + ob), "v"(gB2 + k0) : "memory");
        asm volatile("global_load_async_to_lds_b128 %0, %1, off"
                     :: "v"(laB3 + ob), "v"(gB3 + k0) : "memory");
    };

    v8f acc = {};
    const int NC = D_SZ / BK;                // 64 chunks

    issue(0, 0);
    for (int c = 0; c < NC; ++c) {
        const int buf = c & 1;
        if (c + 1 < NC) {
            issue((c + 1) * BK, buf ^ 1);
            // chunk c complete once only the 5 just-issued loads remain outstanding
            asm volatile("s_wait_asynccnt 5" ::: "memory");
        } else {
            asm volatile("s_wait_asynccnt 0" ::: "memory");
        }
        __syncthreads();   // all waves' chunk-c data visible in LDS

        const float* A  = &lsA[buf][0];
        const float* Bt = &lsB[buf][0];
        #pragma unroll
        for (int kk = 0; kk < BK; kk += 4) {
            const v2f afrag = *(const v2f*)&A[lm * LDSS + kk + koff];
            const v2f bfrag = *(const v2f*)&Bt[(wave * 16 + lm) * LDSS + kk + koff];
            acc = __builtin_amdgcn_wmma_f32_16x16x4_f32(
                /*neg_a=*/false, afrag, /*neg_b=*/false, bfrag,
                /*c_mod=*/(short)0, acc, /*reuse_a=*/false, /*reuse_b=*/false);
        }
        __syncthreads();   // buffer buf is free for iteration c+1's issue
    }

    // ---- epilogue: scale by 1/TEMP and store per C/D layout ----
    // VGPR r: lanes 0-15 -> M=r, lanes 16-31 -> M=r+8; N = lane%16
    const int ncol  = n0 + wave * 16 + lm;
    const int mbase = m0 + (lane >> 4) * 8;
    #pragma unroll
    for (int r = 0; r < 8; ++r)
        logits[(size_t)(mbase + r) * N_SZ + ncol] = acc[r] * INV_TEMP;
}

// ---------------- Kernel 3: per-row loss ----------------
// loss_b = (lmax + log Σ exp(l - lmax)) - (Σ exp(t - tmax) * l) / (Σ exp(t - tmax))
__global__ void __launch_bounds__(256)
row_loss(const float* __restrict__ logits, const float* __restrict__ targets,
         float* __restrict__ rowloss) {
    __shared__ float red[256];
    const int b = blockIdx.x;
    const int t = threadIdx.x;
    const float* lg = logits + (size_t)b * N_SZ;
    const float* tg = targets + (size_t)b * N_SZ;

    float lmax = -1e30f, tmax = -1e30f;
    for (int i = t; i < N_SZ; i += 256) {
        lmax = fmaxf(lmax, lg[i]);
        tmax = fmaxf(tmax, tg[i]);
    }
    red[t] = lmax; __syncthreads();
    for (int off = 128; off > 0; off >>= 1) {
        if (t < off) red[t] = fmaxf(red[t], red[t + off]);
        __syncthreads();
    }
    lmax = red[0]; __syncthreads();
    red[t] = tmax; __syncthreads();
    for (int off = 128; off > 0; off >>= 1) {
        if (t < off) red[t] = fmaxf(red[t], red[t + off]);
        __syncthreads();
    }
    tmax = red[0]; __syncthreads();

    float sl = 0.f, st = 0.f, sd = 0.f;
    for (int i = t; i < N_SZ; i += 256) {
        const float l = lg[i];
        const float e = expf(tg[i] - tmax);
        sl += expf(l - lmax);
        st += e;
        sd += e * l;
    }
    red[t] = sl; __syncthreads();
    for (int off = 128; off > 0; off >>= 1) { if (t < off) red[t] += red[t + off]; __syncthreads(); }
    sl = red[0]; __syncthreads();
    red[t] = st; __syncthreads();
    for (int off = 128; off > 0; off >>= 1) { if (t < off) red[t] += red[t + off]; __syncthreads(); }
    st = red[0]; __syncthreads();
    red[t] = sd; __syncthreads();
    for (int off = 128; off > 0; off >>= 1) { if (t < off) red[t] += red[t + off]; __syncthreads(); }
    sd = red[0];

    if (t == 0)
        rowloss[b] = (lmax + logf(sl)) - sd / st;
}

// ---------------- Kernel 4: mean over rows ----------------
__global__ void final_reduce(const float* __restrict__ rowloss, float* __restrict__ out) {
    __shared__ float red[64];
    const int t = threadIdx.x;
    red[t] = rowloss[t];
    __syncthreads();
    for (int off = 32; off > 0; off >>= 1) {
        if (t < off) red[t] += red[t + off];
        __syncthreads();
    }
    if (t == 0) out[0] = red[0] / (float)B_SZ;
}

extern "C" void kernel_launch(void* const* d_in, const int* in_sizes, int n_in,
                              void* d_out, int out_size, void* d_ws, size_t ws_size,
                              hipStream_t stream) {
    (void)in_sizes; (void)n_in; (void)out_size; (void)ws_size;
    const float* inputs   = (const float*)d_in[0];  // [64, 2048]
    const float* targets  = (const float*)d_in[1];  // [64, 16384]
    /* d_in[2] = cid (unused by the reference loss) */
    const float* features = (const float*)d_in[3];  // [16384, 2048]

    float* ws      = (float*)d_ws;
    float* xn      = ws;                                  // 64*2048
    float* logits  = xn + (size_t)B_SZ * D_SZ;            // 64*16384
    float* rowloss = logits + (size_t)B_SZ * N_SZ;        // 64

    normalize_rows<<<B_SZ, 256, 0, stream>>>(inputs, xn);
    dim3 grid(N_SZ / 128, B_SZ / 16);
    gemm_wmma<<<grid, 256, 0, stream>>>(xn, features, logits);
    row_loss<<<B_SZ, 256, 0, stream>>>(logits, targets, rowloss);
    final_reduce<<<1, 64, 0, stream>>>(rowloss, (float*)d_out);
}